// ts_zscore_65455301591105
// MI455X (gfx1250) — compile-verified
//
#include <hip/hip_runtime.h>

// ---------------------------------------------------------------------------
// Rolling-window z-score: x (64,4096,128) f32, W=30, stride 1 -> (64,4067,128)
//
// MI455X roofline: ~300 MB HBM traffic (1.25x-read input + output) => ~13 us
// @ 23.3 TB/s, bandwidth bound. Strategy:
//   * async global->LDS staging (global_load_async_to_lds_b128 / ASYNCcnt),
//   * window sums via V_WMMA_F32_16X16X4_F32 with a constant banded matrix
//     whose entries are 1/30 (so D = mean / mean-of-squares directly),
//   * epilogue: v_sqrt_f32 + v_rcp_f32, one base pointer per tile with
//     immediate-offset b32 stores (no per-store 64-bit address math).
// ---------------------------------------------------------------------------

#define B_      64
#define T_      4096
#define F_      128
#define W_      30
#define KOUT    4067          // T - W + 1
#define KLAST   4051          // KOUT - 16 (last full 16-tile start)
#define NKT     8             // 16-wide k-tiles per block
#define KCHUNK  (NKT * 16)    // 128 outputs (k) per block
#define ROWS    (KCHUNK + 32) // 160 staged time rows (48 needed by last tile)
#define PITCH   144           // LDS row pitch in floats (16-word pad => no bank conflicts)
#define NF4     (ROWS * (F_ / 4)) // 5120 float4 staging transfers
#define EPS_    1.0e-4f

typedef __attribute__((ext_vector_type(2))) float v2f;
typedef __attribute__((ext_vector_type(8))) float v8f;

// async global->LDS builtin takes (int4 AS1*, int4 AS3*, imm, imm)
typedef int v4i_t __attribute__((vector_size(16)));
typedef __attribute__((address_space(1))) v4i_t g_v4i;
typedef __attribute__((address_space(3))) v4i_t l_v4i;

#ifndef __has_builtin
#define __has_builtin(x) 0
#endif
#if __has_builtin(__builtin_amdgcn_global_load_async_to_lds_b128) && \
    __has_builtin(__builtin_amdgcn_s_wait_asynccnt)
#define HAVE_ASYNC 1
#else
#define HAVE_ASYNC 0
#endif

__global__ __launch_bounds__(256)
void ts_zscore_65455301591105_kernel(const float* __restrict__ x,
                                     float* __restrict__ out)
{
    __shared__ float lds[ROWS * PITCH];   // 92160 B of the 320 KB WGP LDS

    const int tid   = threadIdx.x;
    const int lane  = tid & 31;
    const int wv    = tid >> 5;           // 8 waves: one 16-wide f-tile each
    const int b     = blockIdx.x >> 5;    // 32 k-chunks per batch element
    const int j     = blockIdx.x & 31;
    const int kbase = j * KCHUNK;

    const float* xb = x + (size_t)b * T_ * F_;

    // ---- stage ROWS x 128 f32 into LDS (async global -> LDS, ASYNCcnt) ----
    for (int idx = tid; idx < NF4; idx += 256) {     // exactly 20 iters/thread
        const int row = idx >> 5;                    // 32 float4 per row
        const int q   = (idx & 31) << 2;
        int t = kbase + row;
        t = (t < T_ - 1) ? t : (T_ - 1);             // tail clamp (coeff==0 there)
        const float* g = xb + (size_t)t * F_ + q;
        float*       l = &lds[row * PITCH + q];
#if HAVE_ASYNC
        __builtin_amdgcn_global_load_async_to_lds_b128((g_v4i*)g, (l_v4i*)l, 0, 0);
#else
        *(float4*)l = *(const float4*)g;
#endif
    }
#if HAVE_ASYNC
    __builtin_amdgcn_s_wait_asynccnt(0);
#endif
    __syncthreads();

    // ---- constant banded window matrix A, entries 1/30 (folds the mean) ----
    // A VGPR layout: lanes 0-15 hold M=0..15 with K={0,1}; lanes 16-31 K={2,3}.
    const int   mrow  = lane & 15;
    const int   hi    = lane >> 4;        // 0: low half, 1: high half
    const int   kA    = hi * 2;
    const float inv_w = 1.0f / (float)W_;
    v2f A[12];
#pragma unroll
    for (int c = 0; c < 12; ++c) {
        const int t0 = c * 4 + kA;
        const int t1 = t0 + 1;
        A[c].x = (t0 >= mrow && t0 < mrow + W_) ? inv_w : 0.0f;
        A[c].y = (t1 >= mrow && t1 < mrow + W_) ? inv_w : 0.0f;
    }

    const int f0   = wv * 16;
    const int fcol = f0 + mrow;
    float* const outb = out + (size_t)b * KOUT * F_ + (hi * 8) * F_ + fcol;

    for (int m = 0; m < NKT; ++m) {
        int k0 = kbase + 16 * m;
        if (k0 > KLAST) k0 = KLAST;       // uniform per block; duplicate writes
        const int tbase = k0 - kbase;     // are bit-identical (deterministic)

        v8f s1 = {0.f, 0.f, 0.f, 0.f, 0.f, 0.f, 0.f, 0.f};
        v8f s2 = {0.f, 0.f, 0.f, 0.f, 0.f, 0.f, 0.f, 0.f};

        // B VGPR layout (4x16 f32): v0 lanes0-15 K=0 / lanes16-31 K=1; v1 K=2/3.
        const int bbase = (tbase + hi) * PITCH + fcol;
#pragma unroll
        for (int c = 0; c < 12; ++c) {
            const float bx = lds[bbase + (4 * c) * PITCH];
            const float by = lds[bbase + (4 * c + 2) * PITCH];
            v2f bb; bb.x = bx;      bb.y = by;
            v2f bq; bq.x = bx * bx; bq.y = by * by;
            s1 = __builtin_amdgcn_wmma_f32_16x16x4_f32(false, A[c], false, bb,
                                                       (short)0, s1, false, false);
            s2 = __builtin_amdgcn_wmma_f32_16x16x4_f32(false, A[c], false, bq,
                                                       (short)0, s2, false, false);
        }

        // ---- epilogue: D layout is VGPR r, lane l -> k = k0 + r + 8*hi, f = fcol
        const int lbase = (tbase + hi * 8 + (W_ - 1)) * PITCH + fcol;
        float* const op = outb + (size_t)k0 * F_;    // one u64 add per tile
#pragma unroll
        for (int r = 0; r < 8; ++r) {
            const float mean = s1[r];                // 1/30 folded into A
            const float msq  = s2[r];
            float var = msq - mean * mean;
            var = (var > 0.0f) ? var : 0.0f;
            const float stdev = __builtin_amdgcn_sqrtf(var);
            const float last  = lds[lbase + r * PITCH];
            op[r * F_] = (last - mean) * __builtin_amdgcn_rcpf(stdev + EPS_);
        }
    }
}

extern "C" void kernel_launch(void* const* d_in, const int* in_sizes, int n_in,
                              void* d_out, int out_size, void* d_ws, size_t ws_size,
                              hipStream_t stream)
{
    (void)in_sizes; (void)n_in; (void)out_size; (void)d_ws; (void)ws_size;
    const float* x   = (const float*)d_in[0];
    float*       out = (float*)d_out;
    // 64 batches x 32 k-chunks = 2048 blocks, 256 threads (8 wave32) each.
    ts_zscore_65455301591105_kernel<<<dim3(B_ * 32), dim3(256), 0, stream>>>(x, out);
}